// TensorRingModel_48490180772353
// MI455X (gfx1250) — compile-verified
//
#include <hip/hip_runtime.h>
#include <math.h>

// ---------------- model constants ----------------
#define NQ 20          // qubits
#define NL 4           // layers
#define RK 24          // bond rank
#define TR 48          // 2*RK
#define DD 576         // RK*RK  (transfer-matrix dimension)
#define CORE_ELEMS (RK*2*RK)          // 1152 complex per core
#define ALL_CORE (NQ*CORE_ELEMS)      // 23040

typedef __attribute__((ext_vector_type(8))) float v8f;
typedef __attribute__((ext_vector_type(2))) float v2f;

// ---------------- init cores: identity bonds + small deterministic "noise" ----
__device__ inline unsigned hashu(unsigned x) {
  x ^= x >> 16; x *= 0x7feb352du; x ^= x >> 15; x *= 0x846ca68bu; x ^= x >> 16;
  return x;
}

__global__ void k_init_cores(float2* __restrict__ cores) {
  int idx = blockIdx.x * blockDim.x + threadIdx.x;
  if (idx >= ALL_CORE) return;
  int b = idx % RK;
  int t = idx / RK;
  int s = t & 1; t >>= 1;
  int a = t % RK;
  unsigned h1 = hashu(idx * 2654435761u + 0x9e3779b9u);
  unsigned h2 = hashu(idx * 0x85ebca6bu + 0xc2b2ae35u);
  float u1 = ((h1 >> 8) + 1.0f) * (1.0f / 16777217.0f);
  float u2 = (h2 >> 8) * (1.0f / 16777216.0f);
  float rmag = sqrtf(-2.0f * logf(u1));
  float ang = 6.2831853f * u2;
  float2 c;
  c.x = 1e-2f * rmag * cosf(ang);
  c.y = 1e-2f * rmag * sinf(ang);
  if (s == 0 && a == b) c.x += 1.0f;
  cores[idx] = c;
}

// ---------------- RY layer: new[a,t,b] = sum_s U[t,s] core[a,s,b] ----------
__global__ void k_apply_ry(float2* __restrict__ cores,
                           const float* __restrict__ params, int layer) {
  int idx = blockIdx.x * blockDim.x + threadIdx.x;   // over NQ*RK*RK
  if (idx >= NQ * RK * RK) return;
  int b = idx % RK;
  int a = (idx / RK) % RK;
  int q = idx / (RK * RK);
  float half = 0.5f * params[layer * NQ + q];
  float c = cosf(half), s = sinf(half);
  float2* base = cores + (size_t)q * CORE_ELEMS + (size_t)a * 2 * RK;
  float2 v0 = base[b];
  float2 v1 = base[RK + b];
  float2 n0, n1;
  n0.x = c * v0.x - s * v1.x;  n0.y = c * v0.y - s * v1.y;
  n1.x = s * v0.x + c * v1.x;  n1.y = s * v0.y + c * v1.y;
  base[b] = n0;
  base[RK + b] = n1;
}

// ---------------- CZ gate + truncated SVD (rank 48 -> 24), one workgroup ----
#define SVD_T 128
#define SWEEPS 8

__global__ __launch_bounds__(SVD_T)
void k_cz_svd(float2* __restrict__ cores, int qi, int qj) {
  __shared__ float2 Ms[TR][TR];   // theta as 48x48 matrix (columns rotated)
  __shared__ float2 Vs[TR][TR];   // accumulated right rotations
  __shared__ float sig[TR];
  __shared__ int ord[TR];
  const int tid = threadIdx.x;
  const float2* ci = cores + (size_t)qi * CORE_ELEMS;
  const float2* cj = cores + (size_t)qj * CORE_ELEMS;

  // M[(a,s),(t,c)] = sum_b ci[a,s,b] cj[b,t,c] * (s==1&&t==1 ? -1 : 1)
  for (int e = tid; e < TR * TR; e += SVD_T) {
    int row = e / TR, col = e % TR;
    int a = row >> 1, s = row & 1;
    int t = col / RK, c = col % RK;
    float2 acc = make_float2(0.f, 0.f);
    for (int b = 0; b < RK; ++b) {
      float2 x = ci[(a * 2 + s) * RK + b];
      float2 y = cj[(b * 2 + t) * RK + c];
      acc.x += x.x * y.x - x.y * y.y;
      acc.y += x.x * y.y + x.y * y.x;
    }
    if (s == 1 && t == 1) { acc.x = -acc.x; acc.y = -acc.y; }
    Ms[row][col] = acc;
    Vs[row][col] = make_float2(row == col ? 1.f : 0.f, 0.f);
  }
  __syncthreads();

  // one-sided complex Jacobi: tournament schedule, 24 disjoint pairs/round
  for (int sweep = 0; sweep < SWEEPS; ++sweep) {
    for (int round = 0; round < TR - 1; ++round) {
      if (tid < TR / 2) {
        const int n1 = TR - 1;
        int i, j;
        if (tid == 0) { i = n1; j = round % n1; }
        else { i = (round + tid) % n1; j = (round + n1 - tid) % n1; }
        if (i > j) { int t2 = i; i = j; j = t2; }
        if (i != j) {
          float aa = 0.f, bb = 0.f;
          float2 cc = make_float2(0.f, 0.f);
          for (int r = 0; r < TR; ++r) {
            float2 x = Ms[r][i], y = Ms[r][j];
            aa += x.x * x.x + x.y * x.y;
            bb += y.x * y.x + y.y * y.y;
            cc.x += x.x * y.x + x.y * y.y;   // Re(x^H y)
            cc.y += x.x * y.y - x.y * y.x;   // Im(x^H y)
          }
          float cmag = sqrtf(cc.x * cc.x + cc.y * cc.y);
          if (cmag > 1e-12f * sqrtf(aa * bb) + 1e-30f) {
            float2 ph = make_float2(cc.x / cmag, cc.y / cmag);   // e^{i phi}
            float tau = (bb - aa) / (2.f * cmag);
            float tt = (tau >= 0.f ? 1.f : -1.f) /
                       (fabsf(tau) + sqrtf(1.f + tau * tau));
            float cs = rsqrtf(1.f + tt * tt);
            float sn = tt * cs;
            for (int r = 0; r < TR; ++r) {
              float2 x = Ms[r][i], y = Ms[r][j];
              float2 yp = make_float2(y.x * ph.x + y.y * ph.y,
                                      y.y * ph.x - y.x * ph.y);  // y*conj(ph)
              Ms[r][i] = make_float2(cs * x.x - sn * yp.x, cs * x.y - sn * yp.y);
              Ms[r][j] = make_float2(sn * x.x + cs * yp.x, sn * x.y + cs * yp.y);
              float2 vx = Vs[r][i], vy = Vs[r][j];
              float2 vp = make_float2(vy.x * ph.x + vy.y * ph.y,
                                      vy.y * ph.x - vy.x * ph.y);
              Vs[r][i] = make_float2(cs * vx.x - sn * vp.x, cs * vx.y - sn * vp.y);
              Vs[r][j] = make_float2(sn * vx.x + cs * vp.x, sn * vx.y + cs * vp.y);
            }
          }
        }
      }
      __syncthreads();
    }
  }

  // singular values and top-RK ordering
  if (tid < TR) {
    float nn = 0.f;
    for (int r = 0; r < TR; ++r) {
      float2 x = Ms[r][tid];
      nn += x.x * x.x + x.y * x.y;
    }
    sig[tid] = sqrtf(nn);
  }
  __syncthreads();
  if (tid == 0) {
    for (int k = 0; k < TR; ++k) ord[k] = k;
    for (int k = 0; k < RK; ++k) {
      int best = k;
      for (int m = k + 1; m < TR; ++m)
        if (sig[ord[m]] > sig[ord[best]]) best = m;
      int t2 = ord[k]; ord[k] = ord[best]; ord[best] = t2;
    }
  }
  __syncthreads();

  // ci_new[(a,s),k] = Mcol/sqrt(sigma) ; cj_new[k,(t,c)] = sqrt(sigma)*conj(Vcol)
  float2* co_i = cores + (size_t)qi * CORE_ELEMS;
  float2* co_j = cores + (size_t)qj * CORE_ELEMS;
  for (int e = tid; e < TR * RK; e += SVD_T) {
    int row = e / RK, k = e % RK;     // row = a*2+s
    int col = ord[k];
    float sv = fmaxf(sig[col], 1e-20f);
    float inv = rsqrtf(sv);
    float2 u = Ms[row][col];
    co_i[row * RK + k] = make_float2(u.x * inv, u.y * inv);
  }
  for (int e = tid; e < RK * TR; e += SVD_T) {
    int k = e / TR, tc = e % TR;      // tc = t*RK + c
    int t = tc / RK, c = tc % RK;
    int col = ord[k];
    float sq = sqrtf(fmaxf(sig[col], 0.f));
    float2 v = Vs[tc][col];
    co_j[(k * 2 + t) * RK + c] = make_float2(sq * v.x, -sq * v.y);
  }
}

// ---------------- transfer matrices -----------------------------------------
// E[(b,r),(c,d)] = sum_s z(s) * A[b,s,c] * conj(A[r,s,d])
__global__ void k_build_E(const float2* __restrict__ cores, int q, int zflag,
                          float* __restrict__ Er, float* __restrict__ Ei) {
  int idx = blockIdx.x * blockDim.x + threadIdx.x;
  if (idx >= DD * DD) return;
  int col = idx % DD, rowE = idx / DD;
  int b = rowE / RK, r = rowE % RK;
  int c = col / RK, d = col % RK;
  const float2* A = cores + (size_t)q * CORE_ELEMS;
  float2 acc = make_float2(0.f, 0.f);
#pragma unroll
  for (int s = 0; s < 2; ++s) {
    float z = (zflag && s == 1) ? -1.f : 1.f;
    float2 x = A[(b * 2 + s) * RK + c];
    float2 y = A[(r * 2 + s) * RK + d];
    acc.x += z * (x.x * y.x + x.y * y.y);
    acc.y += z * (x.y * y.x - x.x * y.y);
  }
  Er[idx] = acc.x;
  Ei[idx] = acc.y;
}

__global__ void k_set_identity(float* __restrict__ Mr, float* __restrict__ Mi) {
  int idx = blockIdx.x * blockDim.x + threadIdx.x;
  if (idx >= DD * DD) return;
  int row = idx / DD, col = idx % DD;
  Mr[idx] = (row == col) ? 1.f : 0.f;
  Mi[idx] = 0.f;
}

// ---- TDM: DMA one 64x64 f32 tile (row stride DD) from global into LDS ------
// Builds the D# (group0: 4 SGPRs, group1: 8 SGPRs) per CDNA5 ISA sec 8 and
// issues tensor_load_to_lds (2-group form; tensors up to 2D). TENSORcnt
// tracked; caller waits with __builtin_amdgcn_s_wait_tensorcnt(0).
__device__ inline void tdm_load_tile_64x64(const float* src, unsigned lds_addr) {
  unsigned long long ga = (unsigned long long)(size_t)src;
  unsigned g00 = 1u;                               // count=1, user mode
  unsigned g01 = lds_addr;                         // lds_addr
  unsigned g02 = (unsigned)(ga & 0xffffffffu);     // global_addr[31:0]
  unsigned g03 = ((unsigned)(ga >> 32) & 0x01ffffffu) | 0x80000000u; // +type=2
  unsigned g10 = 2u << 16;                         // data_size=4B, no flags
  unsigned g11 = 576u << 16;                       // tensor_dim0[15:0]=576
  unsigned g12 = 576u << 16;                       // tensor_dim1[15:0]=576
  unsigned g13 = 64u << 16;                        // tile_dim0=64
  unsigned g14 = 64u;                              // tile_dim1=64, tile_dim2=0
  unsigned g15 = 576u;                             // tensor_dim0_stride=576
  unsigned g16 = 0u;
  unsigned g17 = 0u;
  asm volatile(
      "s_mov_b32 s64, %0\n\t"
      "s_mov_b32 s65, %1\n\t"
      "s_mov_b32 s66, %2\n\t"
      "s_mov_b32 s67, %3\n\t"
      "s_mov_b32 s72, %4\n\t"
      "s_mov_b32 s73, %5\n\t"
      "s_mov_b32 s74, %6\n\t"
      "s_mov_b32 s75, %7\n\t"
      "s_mov_b32 s76, %8\n\t"
      "s_mov_b32 s77, %9\n\t"
      "s_mov_b32 s78, %10\n\t"
      "s_mov_b32 s79, %11\n\t"
      "tensor_load_to_lds s[64:67], s[72:79]\n\t"
      :
      : "s"(g00), "s"(g01), "s"(g02), "s"(g03),
        "s"(g10), "s"(g11), "s"(g12), "s"(g13),
        "s"(g14), "s"(g15), "s"(g16), "s"(g17)
      : "s64", "s65", "s66", "s67",
        "s72", "s73", "s74", "s75", "s76", "s77", "s78", "s79", "memory");
}

// ---------------- complex GEMM 576x576x576 via V_WMMA_F32_16X16X4_F32 -------
// split real/imag planes; 8 waves/block arranged 2x4; block tile 32x64.
// B (the E matrix) is staged per K-chunk into LDS by the Tensor Data Mover.
__global__ __launch_bounds__(256)
void k_cgemm(const float* __restrict__ Ar, const float* __restrict__ Ai,
             const float* __restrict__ Br, const float* __restrict__ Bi,
             float* __restrict__ Cr, float* __restrict__ Ci) {
  __shared__ float Bs[2][64][64];         // [plane][k_local][col]  32 KB

  const int wave = threadIdx.x >> 5;      // wave32
  const int lane = threadIdx.x & 31;
  const int wr = wave >> 2;               // 0..1
  const int wc = wave & 3;                // 0..3
  const int row0 = blockIdx.x * 32 + wr * 16;
  const int col0 = blockIdx.y * 64;       // block's 64-wide column panel
  const int mrow = row0 + (lane & 15);    // A rows: both half-waves same M
  const int ncol_l = wc * 16 + (lane & 15);   // column within the LDS panel
  const int khalf = (lane >> 4) * 2;      // K sub-offset: 0 or 2

  v8f crr = {0.f,0.f,0.f,0.f,0.f,0.f,0.f,0.f};
  v8f cii = {0.f,0.f,0.f,0.f,0.f,0.f,0.f,0.f};
  v8f cri = {0.f,0.f,0.f,0.f,0.f,0.f,0.f,0.f};
  v8f cir = {0.f,0.f,0.f,0.f,0.f,0.f,0.f,0.f};

  const unsigned lds0 = (unsigned)(size_t)&Bs[0][0][0];
  const unsigned lds1 = (unsigned)(size_t)&Bs[1][0][0];

  for (int k0 = 0; k0 < DD; k0 += 64) {
    if (wave == 0) {
      // DMA the 64x64 tiles of both E planes into LDS (TDM, TENSORcnt)
      tdm_load_tile_64x64(Br + (size_t)k0 * DD + col0, lds0);
      tdm_load_tile_64x64(Bi + (size_t)k0 * DD + col0, lds1);
      __builtin_amdgcn_s_wait_tensorcnt(0);
    }
    __syncthreads();

    // prefetch next A panel rows while computing this chunk
    if (k0 + 64 < DD) {
      __builtin_prefetch(Ar + (size_t)mrow * DD + k0 + 64, 0, 0);
      __builtin_prefetch(Ai + (size_t)mrow * DD + k0 + 64, 0, 0);
    }

    for (int kk = 0; kk < 64; kk += 4) {
      const int ka = k0 + kk + khalf;       // global K for A frag
      const int kl = kk + khalf;            // local K for B frag
      const float* pA = Ar + (size_t)mrow * DD + ka;
      const float* qA = Ai + (size_t)mrow * DD + ka;
      v2f ar, ai, br, bi;
      ar.x = pA[0]; ar.y = pA[1];
      ai.x = qA[0]; ai.y = qA[1];
      br.x = Bs[0][kl][ncol_l]; br.y = Bs[0][kl + 1][ncol_l];
      bi.x = Bs[1][kl][ncol_l]; bi.y = Bs[1][kl + 1][ncol_l];
      crr = __builtin_amdgcn_wmma_f32_16x16x4_f32(false, ar, false, br,
                                                  (short)0, crr, false, false);
      cii = __builtin_amdgcn_wmma_f32_16x16x4_f32(false, ai, false, bi,
                                                  (short)0, cii, false, false);
      cri = __builtin_amdgcn_wmma_f32_16x16x4_f32(false, ar, false, bi,
                                                  (short)0, cri, false, false);
      cir = __builtin_amdgcn_wmma_f32_16x16x4_f32(false, ai, false, br,
                                                  (short)0, cir, false, false);
    }
    __syncthreads();
  }

  const int colW = col0 + wc * 16 + (lane & 15);
  const int rbase = row0 + ((lane >> 4) << 3);   // +8 for upper half-wave
#pragma unroll
  for (int v = 0; v < 8; ++v) {
    int rr = rbase + v;
    Cr[(size_t)rr * DD + colW] = crr[v] - cii[v];
    Ci[(size_t)rr * DD + colW] = cri[v] + cir[v];
  }
}

// ---------------- trace + final scalar ---------------------------------------
__global__ __launch_bounds__(256)
void k_trace(const float* __restrict__ Mr, float* __restrict__ out, int slot) {
  __shared__ float red[256];
  float acc = 0.f;
  for (int i = threadIdx.x; i < DD; i += 256)
    acc += Mr[(size_t)i * DD + i];
  red[threadIdx.x] = acc;
  __syncthreads();
  for (int s = 128; s > 0; s >>= 1) {
    if (threadIdx.x < s) red[threadIdx.x] += red[threadIdx.x + s];
    __syncthreads();
  }
  if (threadIdx.x == 0) out[slot] = red[0];
}

__global__ void k_final(const float* __restrict__ tr, float* __restrict__ out) {
  out[0] = (0.8f * tr[1] + 0.5f * tr[2]) / tr[0];
}

// ---------------- host driver ------------------------------------------------
static inline int zflag_for(int e, int q) {
  if (e == 1) return (q == 0 || q == 1) ? 1 : 0;   // 0.8 * Z0 Z1
  if (e == 2) return (q == 3) ? 1 : 0;             // 0.5 * Z3
  return 0;                                        // norm
}

extern "C" void kernel_launch(void* const* d_in, const int* in_sizes, int n_in,
                              void* d_out, int out_size, void* d_ws,
                              size_t ws_size, hipStream_t stream) {
  (void)in_sizes; (void)n_in; (void)out_size; (void)ws_size;
  const float* params = (const float*)d_in[0];
  float* out = (float*)d_out;

  char* ws = (char*)d_ws;
  size_t off = 0;
  auto carve = [&](size_t bytes) -> char* {
    char* p = ws + off;
    off += (bytes + 255) & ~(size_t)255;
    return p;
  };
  const size_t plane = (size_t)DD * DD * sizeof(float);   // 1.327 MB
  float2* cores = (float2*)carve(sizeof(float2) * ALL_CORE);
  float* Er = (float*)carve(plane);
  float* Ei = (float*)carve(plane);
  float* M0r = (float*)carve(plane);
  float* M0i = (float*)carve(plane);
  float* M1r = (float*)carve(plane);
  float* M1i = (float*)carve(plane);
  float* traces = (float*)carve(sizeof(float) * 16);

  // 1) initial cores
  k_init_cores<<<(ALL_CORE + 255) / 256, 256, 0, stream>>>(cores);

  // 2) circuit: 4 layers of RY + ring of CZ (each CZ step is a truncated SVD)
  for (int layer = 0; layer < NL; ++layer) {
    k_apply_ry<<<(NQ * RK * RK + 255) / 256, 256, 0, stream>>>(cores, params,
                                                               layer);
    for (int q = 0; q < NQ; ++q)
      k_cz_svd<<<1, SVD_T, 0, stream>>>(cores, q, (q + 1) % NQ);
  }

  // 3) three ring expectation values via transfer-matrix GEMM chains
  const int eg = (DD * DD + 255) / 256;
  dim3 gg(DD / 32, DD / 64);
  for (int e = 0; e < 3; ++e) {
    k_set_identity<<<eg, 256, 0, stream>>>(M0r, M0i);
    float *Arp = M0r, *Aip = M0i, *Crp = M1r, *Cip = M1i;
    for (int q = 0; q < NQ; ++q) {
      k_build_E<<<eg, 256, 0, stream>>>(cores, q, zflag_for(e, q), Er, Ei);
      k_cgemm<<<gg, 256, 0, stream>>>(Arp, Aip, Er, Ei, Crp, Cip);
      float* t;
      t = Arp; Arp = Crp; Crp = t;
      t = Aip; Aip = Cip; Cip = t;
    }
    k_trace<<<1, 256, 0, stream>>>(Arp, traces, e);
  }

  // 4) final scalar: (0.8*tr1 + 0.5*tr2) / norm
  k_final<<<1, 1, 0, stream>>>(traces, out);
}